// AdditiveAttention_64862596104842
// MI455X (gfx1250) — compile-verified
//
#include <hip/hip_runtime.h>
#include <hip/hip_bf16.h>
#include <math.h>

// Problem constants (match reference)
#define B_   8
#define NQ_  256
#define NKV_ 1024
#define DQ_  256
#define DV_  256
#define H_   128
#define NEG_INF_ -1e20f

typedef __attribute__((ext_vector_type(2))) float v2f;
typedef __attribute__((ext_vector_type(8))) float v8f;
typedef __attribute__((ext_vector_type(4))) int   v4i;

// Fast hardware tanh (CDNA5 V_TANH_F32 is a TRANS32 op).
static __device__ inline float fast_tanh(float x) {
#if __has_builtin(__builtin_amdgcn_tanhf)
  return __builtin_amdgcn_tanhf(x);
#else
  float r;
  // v_nop after the trans op: 1 instruction before result use (ISA 7.4 rule 3)
  asm("v_tanh_f32 %0, %1\n\tv_nop" : "=v"(r) : "v"(x));
  return r;
#endif
}

// 16-byte global -> LDS copy. Prefer the CDNA5 async-to-LDS DMA path
// (GLOBAL_LOAD_ASYNC_TO_LDS_B128, ASYNCcnt-tracked, no VGPR round trip);
// fall back to a plain b128 load + ds_store pair.
#if __has_builtin(__builtin_amdgcn_global_load_async_to_lds_b128) && \
    __has_builtin(__builtin_amdgcn_s_wait_asynccnt)
#define ASYNC_LDS_COPY_ 1
typedef __attribute__((address_space(1))) v4i* gv4_ptr_t;
typedef __attribute__((address_space(3))) v4i* lv4_ptr_t;
static __device__ inline void copy16_g2l(const float* g, float* l) {
  // Generic LDS pointer's low 32 bits are the LDS byte offset (aperture rule).
  gv4_ptr_t gp = (gv4_ptr_t)(unsigned long long)g;
  lv4_ptr_t lp = (lv4_ptr_t)(unsigned)(unsigned long long)l;
  __builtin_amdgcn_global_load_async_to_lds_b128(gp, lp, 0, 0);
}
static __device__ inline void copy16_fence_() {
  __builtin_amdgcn_s_wait_asynccnt(0);
}
#else
#define ASYNC_LDS_COPY_ 0
static __device__ inline void copy16_g2l(const float* g, float* l) {
  *(float4*)l = *(const float4*)g;
}
static __device__ inline void copy16_fence_() {}
#endif

// ---------------------------------------------------------------------------
// Kernel 1: projections  qp[m,h] = sum_d Q[m,d]*Wq[h,d]   (m = b*NQ+i, flat)
//                        kp[m,h] = sum_d K[m,d]*Wk[h,d]   (m = b*NKV+j, flat)
// One wave computes one 16x16 (rows x h) tile via v_wmma_f32_16x16x4_f32.
// Wave tiles: qp: 128*8 = 1024, kp: 512*8 = 4096 -> 5120 waves, 4 waves/block.
// ---------------------------------------------------------------------------
__global__ __launch_bounds__(128) void addattn_proj_kernel(
    const float* __restrict__ Q, const float* __restrict__ K,
    const float* __restrict__ Wq, const float* __restrict__ Wk,
    float* __restrict__ qp, float* __restrict__ kp) {
  int wave = blockIdx.x * (blockDim.x >> 5) + (threadIdx.x >> 5);
  int lane = threadIdx.x & 31;

  const float* X;
  const float* W;
  float* Y;
  int tile;
  if (wave < 1024) { X = Q; W = Wq; Y = qp; tile = wave; }
  else             { X = K; W = Wk; Y = kp; tile = wave - 1024; }

  int tm = tile >> 3;          // row tile
  int h0 = (tile & 7) << 4;    // hidden tile * 16
  int m0 = tm << 4;

  int lrow = lane & 15;
  int koff = (lane >> 4) << 1;  // lanes 0-15 -> K=0,1 ; lanes 16-31 -> K=2,3

  const float* arow = X + (size_t)(m0 + lrow) * DQ_ + koff;   // A: 16x4 f32
  const float* brow = W + (size_t)(h0 + lrow) * DQ_ + koff;   // B: 4x16 (Wq^T)

  v8f acc = {};
#pragma unroll 4
  for (int kk = 0; kk < DQ_ / 4; ++kk) {
    v2f a = *(const v2f*)(arow + kk * 4);
    v2f b = *(const v2f*)(brow + kk * 4);
    acc = __builtin_amdgcn_wmma_f32_16x16x4_f32(
        false, a, false, b, (short)0, acc, false, false);
  }

  // D layout: lanes 0-15: M=r,   N=lane   ; lanes 16-31: M=8+r, N=lane-16
  int n = h0 + lrow;
  int mbase = m0 + ((lane >> 4) << 3);
  float* o = Y + n;
#pragma unroll
  for (int r = 0; r < 8; ++r) o[(size_t)(mbase + r) * H_] = acc[r];
}

// ---------------------------------------------------------------------------
// Kernel 2: scores[b,q,k] = sum_h Wv[h]*tanh(qp[b,q,h]+kp[b,k,h]), masked.
// Block = 256 threads, tile = 16 queries x 64 kv.
// Thread (q = t>>4, kvl = t&15) computes 4 scores (kv = kvl + 16j) so the
// sQ/sWv reads are amortized 4x; all LDS reads are ds_load_b128:
//   - sQ  stored [q][132]  : 16B-aligned rows, 2 addrs/wave -> broadcast
//   - sWv                  : single addr/wave -> broadcast
//   - sK  stored [kv][132] : stride 132 = 4 (mod 64); kv = kvl+16j means lane
//     kvl's b128 spans banks {4*kvl+h .. +3} -> 16 lanes tile all 64 banks.
// Staging uses 16B copies (async-to-LDS DMA when available).
// Grid = B * (NQ/16) * (NKV/64) = 8*16*16 = 2048 blocks.
// ---------------------------------------------------------------------------
#define KPAD_ 132
__global__ __launch_bounds__(256) void addattn_score_kernel(
    const float* __restrict__ qp, const float* __restrict__ kp,
    const float* __restrict__ Wv, const int* __restrict__ valid_lens,
    float* __restrict__ scores) {
  __shared__ float sQ[16 * KPAD_];
  __shared__ float sK[64 * KPAD_];
  __shared__ float sWv[H_];

  int bid = blockIdx.x;
  int k0 = (bid & 15) << 6;          // kv tile * 64
  int q0 = ((bid >> 4) & 15) << 4;   // q  tile * 16
  int b  = bid >> 8;

  int t = threadIdx.x;
  if (t < H_ / 4) copy16_g2l(Wv + t * 4, &sWv[t * 4]);

  // Stage q tile (16 x 128 = 512 float4) and k tile (64 x 128 = 2048 float4)
  for (int idx = t; idx < 512; idx += 256) {
    int q = idx >> 5, h = (idx & 31) << 2;
    copy16_g2l(qp + (size_t)(b * NQ_ + q0 + q) * H_ + h, &sQ[q * KPAD_ + h]);
  }
  for (int idx = t; idx < 2048; idx += 256) {
    int kv = idx >> 5, h = (idx & 31) << 2;
    copy16_g2l(kp + (size_t)(b * NKV_ + k0 + kv) * H_ + h, &sK[kv * KPAD_ + h]);
  }
  copy16_fence_();   // wait this wave's ASYNCcnt (no-op on fallback path)
  __syncthreads();

  int q   = t >> 4;      // 0..15
  int kvl = t & 15;      // 0..15
  int vlen = valid_lens[b];

  const float* qrow = &sQ[q * KPAD_];
  const float* kr0 = &sK[(kvl)      * KPAD_];
  const float* kr1 = &sK[(kvl + 16) * KPAD_];
  const float* kr2 = &sK[(kvl + 32) * KPAD_];
  const float* kr3 = &sK[(kvl + 48) * KPAD_];

  float s0 = 0.f, s1 = 0.f, s2 = 0.f, s3 = 0.f;
#pragma unroll 4
  for (int h = 0; h < H_; h += 4) {
    float4 qv = *(const float4*)(qrow + h);   // ds_load_b128, broadcast
    float4 wv = *(const float4*)(sWv + h);    // ds_load_b128, broadcast
    float4 a  = *(const float4*)(kr0 + h);    // ds_load_b128, conflict-free
    float4 bb = *(const float4*)(kr1 + h);
    float4 c  = *(const float4*)(kr2 + h);
    float4 d  = *(const float4*)(kr3 + h);
    s0 = fmaf(wv.x, fast_tanh(qv.x + a.x), s0);
    s1 = fmaf(wv.x, fast_tanh(qv.x + bb.x), s1);
    s2 = fmaf(wv.x, fast_tanh(qv.x + c.x), s2);
    s3 = fmaf(wv.x, fast_tanh(qv.x + d.x), s3);
    s0 = fmaf(wv.y, fast_tanh(qv.y + a.y), s0);
    s1 = fmaf(wv.y, fast_tanh(qv.y + bb.y), s1);
    s2 = fmaf(wv.y, fast_tanh(qv.y + c.y), s2);
    s3 = fmaf(wv.y, fast_tanh(qv.y + d.y), s3);
    s0 = fmaf(wv.z, fast_tanh(qv.z + a.z), s0);
    s1 = fmaf(wv.z, fast_tanh(qv.z + bb.z), s1);
    s2 = fmaf(wv.z, fast_tanh(qv.z + c.z), s2);
    s3 = fmaf(wv.z, fast_tanh(qv.z + d.z), s3);
    s0 = fmaf(wv.w, fast_tanh(qv.w + a.w), s0);
    s1 = fmaf(wv.w, fast_tanh(qv.w + bb.w), s1);
    s2 = fmaf(wv.w, fast_tanh(qv.w + c.w), s2);
    s3 = fmaf(wv.w, fast_tanh(qv.w + d.w), s3);
  }

  float* srow = scores + (size_t)(b * NQ_ + q0 + q) * NKV_ + k0;
  // lanes 0-15 write consecutive kv for each j -> coalesced 64B stores
  srow[kvl]      = (k0 + kvl      >= vlen) ? NEG_INF_ : s0;
  srow[kvl + 16] = (k0 + kvl + 16 >= vlen) ? NEG_INF_ : s1;
  srow[kvl + 32] = (k0 + kvl + 32 >= vlen) ? NEG_INF_ : s2;
  srow[kvl + 48] = (k0 + kvl + 48 >= vlen) ? NEG_INF_ : s3;
}

// ---------------------------------------------------------------------------
// Kernel 3: in-place softmax over the NKV=1024 axis. One block per (b,q) row.
// ---------------------------------------------------------------------------
__global__ __launch_bounds__(256) void addattn_softmax_kernel(
    float* __restrict__ scores) {
  __shared__ float red[256];
  float* s = scores + (size_t)blockIdx.x * NKV_;
  int t = threadIdx.x;

  float4 v = *(const float4*)(s + t * 4);
  float m = fmaxf(fmaxf(v.x, v.y), fmaxf(v.z, v.w));
  red[t] = m;
  __syncthreads();
  for (int off = 128; off > 0; off >>= 1) {
    if (t < off) red[t] = fmaxf(red[t], red[t + off]);
    __syncthreads();
  }
  m = red[0];
  __syncthreads();

  float4 e;
  e.x = __expf(v.x - m); e.y = __expf(v.y - m);
  e.z = __expf(v.z - m); e.w = __expf(v.w - m);
  red[t] = e.x + e.y + e.z + e.w;
  __syncthreads();
  for (int off = 128; off > 0; off >>= 1) {
    if (t < off) red[t] += red[t + off];
    __syncthreads();
  }
  float inv = 1.0f / red[0];
  e.x *= inv; e.y *= inv; e.z *= inv; e.w *= inv;
  *(float4*)(s + t * 4) = e;
}

// ---------------------------------------------------------------------------
// Kernel 4: out[b,q,v] = sum_k attn[b,q,k] * V[b,k,v] via WMMA f32 16x16x4.
// One wave computes a 16(m) x 32(v) strip = 2 WMMA tiles sharing the A
// fragment (halves attn-side L2 traffic, 2x WMMA ILP).
// Waves: (2048/16) * (256/32) = 1024 -> 256 blocks x 4 waves.
// ---------------------------------------------------------------------------
__global__ __launch_bounds__(128) void addattn_av_kernel(
    const float* __restrict__ attn, const float* __restrict__ V,
    float* __restrict__ out) {
  int wave = blockIdx.x * (blockDim.x >> 5) + (threadIdx.x >> 5);
  int lane = threadIdx.x & 31;

  int m0 = (wave >> 3) << 4;   // query-row tile (flat over b*NQ)
  int v0 = (wave & 7) << 5;    // value-dim strip of 32
  int b  = m0 >> 8;            // 256 rows per batch, tiles never straddle

  int lrow = lane & 15;
  int koff = (lane >> 4) << 1;

  const float* arow  = attn + (size_t)(m0 + lrow) * NKV_ + koff;
  const float* bcol0 = V + (size_t)b * NKV_ * DV_ + v0 + lrow;        // N tile 0
  const float* bcol1 = bcol0 + 16;                                     // N tile 1

  v8f acc0 = {};
  v8f acc1 = {};
#pragma unroll 4
  for (int kk = 0; kk < NKV_ / 4; ++kk) {
    v2f a = *(const v2f*)(arow + kk * 4);        // A[M][4kk+koff .. +1]
    size_t r0 = (size_t)(kk * 4 + koff) * DV_;
    size_t r1 = r0 + DV_;
    v2f b0, b1;
    b0.x = bcol0[r0]; b0.y = bcol0[r1];
    b1.x = bcol1[r0]; b1.y = bcol1[r1];
    acc0 = __builtin_amdgcn_wmma_f32_16x16x4_f32(
        false, a, false, b0, (short)0, acc0, false, false);
    acc1 = __builtin_amdgcn_wmma_f32_16x16x4_f32(
        false, a, false, b1, (short)0, acc1, false, false);
  }

  int n = v0 + lrow;
  int mbase = m0 + ((lane >> 4) << 3);
  float* o0 = out + n;
  float* o1 = out + n + 16;
#pragma unroll
  for (int r = 0; r < 8; ++r) {
    o0[(size_t)(mbase + r) * DV_] = acc0[r];
    o1[(size_t)(mbase + r) * DV_] = acc1[r];
  }
}

// ---------------------------------------------------------------------------
extern "C" void kernel_launch(void* const* d_in, const int* in_sizes, int n_in,
                              void* d_out, int out_size, void* d_ws, size_t ws_size,
                              hipStream_t stream) {
  const float* Q  = (const float*)d_in[0];   // (B, NQ, DQ)
  const float* K  = (const float*)d_in[1];   // (B, NKV, DK)
  const float* V  = (const float*)d_in[2];   // (B, NKV, DV)
  const int*   vl = (const int*)  d_in[3];   // (B,)
  const float* Wq = (const float*)d_in[4];   // (H, DQ)
  const float* Wk = (const float*)d_in[5];   // (H, DK)
  const float* Wv = (const float*)d_in[6];   // (H,)
  float* out = (float*)d_out;                // (B, NQ, DV)

  float* wsf   = (float*)d_ws;
  float* qp    = wsf;                                   // B*NQ*H    = 262144 f
  float* kp    = wsf + (size_t)B_ * NQ_ * H_;           // B*NKV*H   = 1048576 f
  float* score = kp  + (size_t)B_ * NKV_ * H_;          // B*NQ*NKV  = 2097152 f

  // 1) projections: 5120 wave-tiles, 4 waves/block
  addattn_proj_kernel<<<1280, 128, 0, stream>>>(Q, K, Wq, Wk, qp, kp);
  // 2) tanh scores + mask: 2048 blocks x 256 threads
  addattn_score_kernel<<<2048, 256, 0, stream>>>(qp, kp, Wv, vl, score);
  // 3) row softmax: one block per (b,q) row
  addattn_softmax_kernel<<<B_ * NQ_, 256, 0, stream>>>(score);
  // 4) attn @ V: 1024 wave-strips, 4 waves/block
  addattn_av_kernel<<<256, 128, 0, stream>>>(score, V, out);
}